// DeepSN_29695403884985
// MI455X (gfx1250) — compile-verified
//
#include <hip/hip_runtime.h>
#include <cstdint>

#define NINPUT 256
#define NFEAT  128
#define NITER  3
#define EPSF   1e-8f

typedef float v2f __attribute__((ext_vector_type(2)));
typedef float v8f __attribute__((ext_vector_type(8)));

__device__ __forceinline__ float sigmoidf(float v) {
    return 1.0f / (1.0f + expf(-v));
}

// ---------------------------------------------------------------- degree
__global__ __launch_bounds__(256) void deg_kernel(const float* __restrict__ vals,
                                                  const int* __restrict__ src,
                                                  float* __restrict__ deg, int nedges) {
    int e = blockIdx.x * 256 + threadIdx.x;
    if (e < nedges) atomicAdd(&deg[src[e]], vals[e]);
}

__global__ __launch_bounds__(256) void dinv_kernel(const float* __restrict__ deg,
                                                   float* __restrict__ dinv, int nnodes) {
    int i = blockIdx.x * 256 + threadIdx.x;
    if (i < nnodes) {
        float d = deg[i];
        dinv[i] = (d > 0.0f) ? rsqrtf(fmaxf(d, 1.0f)) : 0.0f;
    }
}

__global__ __launch_bounds__(256) void vnorm_kernel(const float* __restrict__ vals,
                                                    const int* __restrict__ src,
                                                    const int* __restrict__ dst,
                                                    const float* __restrict__ dinv,
                                                    float* __restrict__ vnorm, int nedges) {
    int e = blockIdx.x * 256 + threadIdx.x;
    if (e < nedges) vnorm[e] = vals[e] * dinv[src[e]] * dinv[dst[e]];
}

__global__ __launch_bounds__(256) void gate_kernel(const float* __restrict__ vals,
                                                   const int* __restrict__ dst,
                                                   const float* __restrict__ ya,
                                                   float* __restrict__ gate, int nedges) {
    int e = blockIdx.x * 256 + threadIdx.x;
    if (e < nedges) gate[e] = vals[e] * (2.0f * ya[dst[e]] - 1.0f);
}

// ------------------------------------------------- GEMM: h = x @ Wt + bt
// fp32 WMMA 16x16x4. One wave computes a 16-row strip x 128 cols.
// A 16x4 f32 fragment: lane L -> M = L&15, K = (L>>4)*2 .. +1  (2 VGPRs)
// B 4x16  f32 fragment: lane L -> N = L&15, K = (L>>4)*2 .. +1 (2 VGPRs)
// C/D: VGPR r -> M = r + 8*(L>>4), N = L&15
__global__ __launch_bounds__(256) void gemm_wmma(const float* __restrict__ x,
                                                 const float* __restrict__ Wt,
                                                 const float* __restrict__ bt,
                                                 float* __restrict__ h, int nrows) {
    const int lane = threadIdx.x & 31;
    const int wave = threadIdx.x >> 5;
    const int rowBase = blockIdx.x * 128 + wave * 16;   // uniform per wave
    if (rowBase >= nrows) return;                       // wave-uniform exit

    const int m  = lane & 15;
    const int kb = (lane >> 4) * 2;

    int row = rowBase + m;
    if (row >= nrows) row = nrows - 1;                  // select, no divergence
    const float* __restrict__ arow = x + (size_t)row * NINPUT;

    v8f acc[8] = {};

    for (int k0 = 0; k0 < NINPUT; k0 += 4) {
        v2f a = *(const v2f*)(arow + k0 + kb);
        const float* __restrict__ brow = Wt + (size_t)(k0 + kb) * NFEAT + m; // m == n
#pragma unroll
        for (int c = 0; c < 8; ++c) {
            v2f b;
            b.x = brow[c * 16];
            b.y = brow[c * 16 + NFEAT];
            acc[c] = __builtin_amdgcn_wmma_f32_16x16x4_f32(
                false, a, false, b, (short)0, acc[c], false, false);
        }
    }

    const int mb = (lane >> 4) * 8;
#pragma unroll
    for (int c = 0; c < 8; ++c) {
        int col = c * 16 + (lane & 15);
        float bias = bt[col];
#pragma unroll
        for (int r = 0; r < 8; ++r) {
            int rr = rowBase + mb + r;
            if (rr < nrows) h[(size_t)rr * NFEAT + col] = acc[c][r] + bias;
        }
    }
}

// ------------------------------------------ SpMM: out[src] += v * x[dst]
// one wave per edge, 4 features per lane
__global__ __launch_bounds__(256) void spmm_scatter(const float* __restrict__ vals,
                                                    const int* __restrict__ src,
                                                    const int* __restrict__ dst,
                                                    const float* __restrict__ xin,
                                                    float* __restrict__ out, int nedges) {
    int t = blockIdx.x * 256 + threadIdx.x;
    int e = t >> 5;
    if (e >= nedges) return;
    int lane = t & 31;
    float v = vals[e];
    int d = dst[e];
    int s = src[e];
    const float4 xv = *(const float4*)(xin + (size_t)d * NFEAT + lane * 4);
    float* op = out + (size_t)s * NFEAT + lane * 4;
    atomicAdd(op + 0, v * xv.x);
    atomicAdd(op + 1, v * xv.y);
    atomicAdd(op + 2, v * xv.z);
    atomicAdd(op + 3, v * xv.w);
}

// ------------------------- pointwise 1: x = x - spmm; x += sigmoid(...)
__global__ __launch_bounds__(256) void pw1_kernel(float* __restrict__ h,
                                                  const float* __restrict__ tmp,
                                                  const float* __restrict__ phi1,
                                                  const float* __restrict__ kap1,
                                                  const float* __restrict__ beta,
                                                  size_t total) {
    size_t i = (size_t)blockIdx.x * 256 + threadIdx.x;
    if (i >= total) return;
    float sb = sigmoidf(beta[0]);
    float xv = h[i] - tmp[i];
    float t  = sb * (phi1[i] * xv) / (kap1[i] + xv + EPSF);
    h[i] = xv + sigmoidf(t);
}

// -------- pointwise 2 + ELU + row-mean(sigmoid) fused: one node per block
__global__ __launch_bounds__(128) void upd2_kernel(float* __restrict__ h,
                                                   const float* __restrict__ xd,
                                                   const float* __restrict__ phi2,
                                                   const float* __restrict__ kap2,
                                                   const float* __restrict__ gma,
                                                   float* __restrict__ ya, int nnodes) {
    int node = blockIdx.x;
    if (node >= nnodes) return;
    int f = threadIdx.x;
    size_t i = (size_t)node * NFEAT + f;

    float sg = sigmoidf(gma[0]);
    float d  = xd[i];
    float t  = sg * (phi2[i] * d) / (kap2[i] + fabsf(d) + EPSF);
    float xv = h[i] + sigmoidf(t);
    xv = (xv > 0.0f) ? xv : (expf(xv) - 1.0f);      // ELU (alpha=1)
    h[i] = xv;

    float s = sigmoidf(xv);
#pragma unroll
    for (int off = 16; off > 0; off >>= 1)
        s += __shfl_xor(s, off, 32);                // wave32 reduction

    __shared__ float red[4];
    if ((f & 31) == 0) red[f >> 5] = s;
    __syncthreads();
    if (f == 0)
        ya[node] = (red[0] + red[1] + red[2] + red[3]) * (1.0f / (float)NFEAT);
}

// ---------------------------------------------------------------- launch
extern "C" void kernel_launch(void* const* d_in, const int* in_sizes, int n_in,
                              void* d_out, int out_size, void* d_ws, size_t ws_size,
                              hipStream_t stream) {
    const float* x    = (const float*)d_in[0];
    const float* Wt   = (const float*)d_in[1];
    const float* bt   = (const float*)d_in[2];
    const float* phi1 = (const float*)d_in[3];
    const float* phi2 = (const float*)d_in[4];
    const float* kap1 = (const float*)d_in[5];
    const float* kap2 = (const float*)d_in[6];
    const float* beta = (const float*)d_in[7];
    const float* gma  = (const float*)d_in[8];
    const float* adj  = (const float*)d_in[9];
    const float* y_i  = (const float*)d_in[10];
    const int*   esrc = (const int*)d_in[11];
    const int*   edst = (const int*)d_in[12];
    // d_in[13] is n == NITER == 3 (compile-time constant in the reference)
    (void)n_in; (void)out_size; (void)ws_size;

    const int Nn = in_sizes[10];   // y_i: N
    const int Ee = in_sizes[9];    // adj_vals: E

    char* w = (char*)d_ws;
    size_t off = 0;
    auto carve = [&](size_t bytes) -> void* {
        void* p = w + off;
        off += (bytes + 255) & ~(size_t)255;
        return p;
    };
    float* deg   = (float*)carve((size_t)Nn * 4);
    float* dinv  = (float*)carve((size_t)Nn * 4);
    float* vnorm = (float*)carve((size_t)Ee * 4);
    float* gate  = (float*)carve((size_t)Ee * 4);
    float* ya    = (float*)carve((size_t)Nn * 4);
    float* h     = (float*)carve((size_t)Nn * NFEAT * 4);
    float* tmp   = (float*)carve((size_t)Nn * NFEAT * 4);

    const int eb = (Ee + 255) / 256;
    const int nb = (Nn + 255) / 256;
    const size_t total = (size_t)Nn * NFEAT;
    const int tb = (int)((total + 255) / 256);
    const int sb = (int)(((size_t)Ee * 32 + 255) / 256);

    // normalized adjacency values
    hipMemsetAsync(deg, 0, (size_t)Nn * 4, stream);
    deg_kernel<<<eb, 256, 0, stream>>>(adj, esrc, deg, Ee);
    dinv_kernel<<<nb, 256, 0, stream>>>(deg, dinv, Nn);
    vnorm_kernel<<<eb, 256, 0, stream>>>(adj, esrc, edst, dinv, vnorm, Ee);

    // h = x @ Wt + bt   (fp32 WMMA)
    gemm_wmma<<<(Nn + 127) / 128, 256, 0, stream>>>(x, Wt, bt, h, Nn);

    hipMemcpyAsync(ya, y_i, (size_t)Nn * 4, hipMemcpyDeviceToDevice, stream);

    for (int it = 0; it < NITER; ++it) {
        hipMemsetAsync(tmp, 0, total * 4, stream);
        spmm_scatter<<<sb, 256, 0, stream>>>(vnorm, esrc, edst, h, tmp, Ee);
        pw1_kernel<<<tb, 256, 0, stream>>>(h, tmp, phi1, kap1, beta, total);

        gate_kernel<<<eb, 256, 0, stream>>>(adj, edst, ya, gate, Ee);
        hipMemsetAsync(tmp, 0, total * 4, stream);
        spmm_scatter<<<sb, 256, 0, stream>>>(gate, esrc, edst, h, tmp, Ee);
        upd2_kernel<<<Nn, 128, 0, stream>>>(h, tmp, phi2, kap2, gma, ya, Nn);
    }

    hipMemcpyAsync(d_out, ya, (size_t)Nn * 4, hipMemcpyDeviceToDevice, stream);
}